// MambaStack_14748917695058
// MI455X (gfx1250) — compile-verified
//
#include <hip/hip_runtime.h>
#include <hip/hip_bf16.h>
#include <cstdint>

// ---------------- Model constants ----------------
#define BB   2
#define TT   1024
#define DM   1024
#define DS   16
#define DC   4
#define DI   2048          // EXPAND * DM
#define DTR  64            // dt_rank
#define XD   96            // DT_RANK + 2*D_STATE
#define ROWS (BB * TT)     // 2048

typedef __attribute__((ext_vector_type(2))) float v2f;
typedef __attribute__((ext_vector_type(8))) float v8f;

// ---------------- CDNA5 async global->LDS helpers ----------------
__device__ __forceinline__ void async_load_b128(uint32_t lds_addr, const void* gptr)
{
    // GLOBAL_LOAD_ASYNC_TO_LDS_B128: VDST = LDS byte address, VADDR = 64-bit global addr
    asm volatile("global_load_async_to_lds_b128 %0, %1, off"
                 :
                 : "v"(lds_addr), "v"((uint64_t)(uintptr_t)gptr)
                 : "memory");
}

__device__ __forceinline__ void wait_async0()
{
#if __has_builtin(__builtin_amdgcn_s_wait_asynccnt)
    __builtin_amdgcn_s_wait_asynccnt(0);
#else
    asm volatile("s_wait_asynccnt 0" ::: "memory");
#endif
}

// ---------------- GEMM: C[M,N] = A[M,K](lda) * W[N,K]^T, fp32 WMMA ----------------
// Workgroup tile 64x128 (8 waves, 2x4), wave tile 32x32 (4 accumulators).
// Double-buffered async global->LDS staging, one barrier per K-chunk.
// epilogue: 0 = none, 1 = softplus(x + bias[col])
#define TM  64
#define TN  128
#define KC  16
#define KCP 20   // LDS row stride in floats: 80B keeps float4 16B-aligned, conflict-free

__global__ __launch_bounds__(256)
void gemm_rt_wmma(const float* __restrict__ A, int lda,
                  const float* __restrict__ W,
                  float* __restrict__ C,
                  int M, int N, int K,
                  const float* __restrict__ bias, int epilogue)
{
    __shared__ float As[2][TM][KCP];
    __shared__ float Ws[2][TN][KCP];

    const int tid  = threadIdx.x;
    const int wave = tid >> 5;
    const int lane = tid & 31;
    const int bm = blockIdx.x * TM;
    const int bn = blockIdx.y * TN;

    const int wm = (wave >> 2) * 32;   // 0,32
    const int wn = (wave & 3) * 32;    // 0,32,64,96

    const int lr  = lane & 15;         // row/col within 16-tile
    const int kk2 = (lane >> 4) * 2;   // K sub-pair select (0 or 2)

    v8f c00 = {}, c01 = {}, c10 = {}, c11 = {};

    // loader mapping: thread t -> (row = t>>2, float4 slot q = t&3)
    const int lrow = tid >> 2;          // 0..63
    const int lq   = (tid & 3) * 4;     // 0,4,8,12

    // W source rows; clamp OOB rows to a valid address (their columns are never stored)
    const int wr0 = bn + lrow;
    const int wr1 = bn + lrow + 64;
    const int wr0c = (wr0 < N) ? wr0 : (N - 1);
    const int wr1c = (wr1 < N) ? wr1 : (N - 1);

    const int nchunks = K / KC;

    auto issue = [&](int chunk, int buf) {
        const int k0 = chunk * KC;
        async_load_b128((uint32_t)(uintptr_t)&As[buf][lrow][lq],
                        &A[(size_t)(bm + lrow) * lda + k0 + lq]);
        async_load_b128((uint32_t)(uintptr_t)&Ws[buf][lrow][lq],
                        &W[(size_t)wr0c * K + k0 + lq]);
        async_load_b128((uint32_t)(uintptr_t)&Ws[buf][lrow + 64][lq],
                        &W[(size_t)wr1c * K + k0 + lq]);
    };

    issue(0, 0);
    wait_async0();
    __syncthreads();

    for (int chunk = 0; chunk < nchunks; ++chunk) {
        const int cur = chunk & 1;
        if (chunk + 1 < nchunks) issue(chunk + 1, cur ^ 1);

#pragma unroll
        for (int kk = 0; kk < KC; kk += 4) {
            v2f a0, a1, b0, b1;
            // A 16x4 layout: lanes 0-15 K={kk,kk+1}, lanes 16-31 K={kk+2,kk+3}
            a0.x = As[cur][wm + lr][kk + kk2];
            a0.y = As[cur][wm + lr][kk + kk2 + 1];
            a1.x = As[cur][wm + 16 + lr][kk + kk2];
            a1.y = As[cur][wm + 16 + lr][kk + kk2 + 1];
            // B 4x16 layout mirrors A's K split; B[k][n] = W[n][k]
            b0.x = Ws[cur][wn + lr][kk + kk2];
            b0.y = Ws[cur][wn + lr][kk + kk2 + 1];
            b1.x = Ws[cur][wn + 16 + lr][kk + kk2];
            b1.y = Ws[cur][wn + 16 + lr][kk + kk2 + 1];
            c00 = __builtin_amdgcn_wmma_f32_16x16x4_f32(false, a0, false, b0, (short)0, c00, false, false);
            c01 = __builtin_amdgcn_wmma_f32_16x16x4_f32(false, a0, false, b1, (short)0, c01, false, false);
            c10 = __builtin_amdgcn_wmma_f32_16x16x4_f32(false, a1, false, b0, (short)0, c10, false, false);
            c11 = __builtin_amdgcn_wmma_f32_16x16x4_f32(false, a1, false, b1, (short)0, c11, false, false);
        }

        wait_async0();
        __syncthreads();
    }

    // C layout: VGPR j -> row = j + 8*(lane>>4), col = lane&15
    const int rowT = bm + wm + (lane >> 4) * 8;
    const int colL = bn + wn + lr;

    auto store_tile = [&](const v8f& cc, int rbase, int col) {
        if (col >= N) return;
        float bv = 0.0f;
        if (epilogue == 1) bv = bias[col];
#pragma unroll
        for (int j = 0; j < 8; ++j) {
            float v = cc[j];
            if (epilogue == 1) {
                v += bv;
                v = (v > 20.0f) ? v : log1pf(__expf(v));
            }
            C[(size_t)(rbase + j) * N + col] = v;
        }
    };

    store_tile(c00, rowT,      colL);
    store_tile(c01, rowT,      colL + 16);
    store_tile(c10, rowT + 16, colL);
    store_tile(c11, rowT + 16, colL + 16);
}

// ---------------- Causal depthwise conv (k=4) + SiLU ----------------
__global__ __launch_bounds__(256)
void conv_silu_kernel(const float* __restrict__ xz,
                      const float* __restrict__ cw,
                      const float* __restrict__ cb,
                      float* __restrict__ u)
{
    const int idx = blockIdx.x * 256 + threadIdx.x;   // over ROWS*DI
    const int d  = idx & (DI - 1);
    const int bt = idx >> 11;                          // DI = 2^11
    const int t  = bt & (TT - 1);
    const int b  = bt >> 10;

    float acc = cb[d];
#pragma unroll
    for (int k = 0; k < DC; ++k) {
        const int tt = t - (DC - 1) + k;
        if (tt >= 0) {
            acc += xz[((size_t)(b * TT + tt)) * (2 * DI) + d] * cw[d * DC + k];
        }
    }
    u[idx] = acc / (1.0f + __expf(-acc));   // silu
}

// ---------------- Selective scan: one thread per (b, d, n) ----------------
// Fuses: h recurrence, y = <h, C>, + u*Dskip, * silu(z)
__global__ __launch_bounds__(256)
void scan_kernel(const float* __restrict__ dtf,     // [ROWS, DI] softplus'd
                 const float* __restrict__ xdbl,    // [ROWS, XD] (B at 64, C at 80)
                 const float* __restrict__ u,       // [ROWS, DI]
                 const float* __restrict__ xz,      // [ROWS, 2*DI] (z at +DI)
                 const float* __restrict__ A_log,   // [DI, DS]
                 const float* __restrict__ Dskip,   // [DI]
                 float* __restrict__ ys)            // [ROWS, DI]
{
    const int g = blockIdx.x * 256 + threadIdx.x;   // 0 .. BB*DI*DS-1
    const int n = g & (DS - 1);
    const int d = (g >> 4) & (DI - 1);
    const int b = g >> 15;

    const float Aval = -__expf(A_log[d * DS + n]);
    const float Dv   = Dskip[d];

    float h = 0.0f;
    const size_t rowBase = (size_t)b * TT;

    for (int t = 0; t < TT; ++t) {
        const size_t row = rowBase + t;
        const float dt_t = dtf[row * DI + d];
        const float u_t  = u[row * DI + d];
        const float B_t  = xdbl[row * XD + DTR + n];
        const float C_t  = xdbl[row * XD + DTR + DS + n];

        const float dA = __expf(dt_t * Aval);
        h = dA * h + (dt_t * u_t) * B_t;

        float s = h * C_t;
        // reduce over the 16 state lanes (stays inside 16-lane group, wave32)
        s += __shfl_xor(s, 1);
        s += __shfl_xor(s, 2);
        s += __shfl_xor(s, 4);
        s += __shfl_xor(s, 8);

        if (n == 0) {
            const float z   = xz[row * (2 * DI) + DI + d];
            const float sil = z / (1.0f + __expf(-z));
            ys[row * DI + d] = (s + u_t * Dv) * sil;
        }
    }
}

// ---------------- LayerNorm over last dim (1024) + residual ----------------
__global__ __launch_bounds__(256)
void ln_residual_kernel(const float* __restrict__ h,
                        const float* __restrict__ resid,
                        const float* __restrict__ w,
                        const float* __restrict__ bb,
                        float* __restrict__ out)
{
    const int row = blockIdx.x;
    const int tid = threadIdx.x;

    const float4 v = reinterpret_cast<const float4*>(h + (size_t)row * DM)[tid];
    float s1 = v.x + v.y + v.z + v.w;
    float s2 = v.x * v.x + v.y * v.y + v.z * v.z + v.w * v.w;

#pragma unroll
    for (int m = 16; m >= 1; m >>= 1) {
        s1 += __shfl_xor(s1, m);
        s2 += __shfl_xor(s2, m);
    }

    __shared__ float r1[8], r2[8];
    __shared__ float mu_s, rstd_s;
    if ((tid & 31) == 0) { r1[tid >> 5] = s1; r2[tid >> 5] = s2; }
    __syncthreads();
    if (tid == 0) {
        float a = 0.0f, q = 0.0f;
#pragma unroll
        for (int i = 0; i < 8; ++i) { a += r1[i]; q += r2[i]; }
        const float mu  = a * (1.0f / (float)DM);
        const float var = q * (1.0f / (float)DM) - mu * mu;
        mu_s   = mu;
        rstd_s = rsqrtf(var + 1e-5f);
    }
    __syncthreads();
    const float mu = mu_s, rstd = rstd_s;

    const float4 rv = reinterpret_cast<const float4*>(resid + (size_t)row * DM)[tid];
    const float4 wv = reinterpret_cast<const float4*>(w)[tid];
    const float4 bv = reinterpret_cast<const float4*>(bb)[tid];

    float4 o;
    o.x = (v.x - mu) * rstd * wv.x + bv.x + rv.x;
    o.y = (v.y - mu) * rstd * wv.y + bv.y + rv.y;
    o.z = (v.z - mu) * rstd * wv.z + bv.z + rv.z;
    o.w = (v.w - mu) * rstd * wv.w + bv.w + rv.w;
    reinterpret_cast<float4*>(out + (size_t)row * DM)[tid] = o;
}

// ---------------- Host orchestration ----------------
extern "C" void kernel_launch(void* const* d_in, const int* in_sizes, int n_in,
                              void* d_out, int out_size, void* d_ws, size_t ws_size,
                              hipStream_t stream)
{
    (void)in_sizes; (void)n_in; (void)out_size; (void)ws_size;

    const float* x      = (const float*)d_in[0];
    const float* in_w   = (const float*)d_in[1];   // [L, 2*DI, DM]
    const float* conv_w = (const float*)d_in[2];   // [L, DI, DC]
    const float* conv_b = (const float*)d_in[3];   // [L, DI]
    const float* xp_w   = (const float*)d_in[4];   // [L, XD, DI]
    const float* dt_w   = (const float*)d_in[5];   // [L, DI, DTR]
    const float* dt_b   = (const float*)d_in[6];   // [L, DI]
    const float* A_log  = (const float*)d_in[7];   // [L, DI, DS]
    const float* Dsk    = (const float*)d_in[8];   // [L, DI]
    const float* out_w  = (const float*)d_in[9];   // [L, DM, DI]
    const float* ln_w   = (const float*)d_in[10];  // [L, DM]
    const float* ln_b   = (const float*)d_in[11];  // [L, DM]

    float* ws   = (float*)d_ws;
    float* xz   = ws;                                // [ROWS, 2*DI]
    float* u    = xz   + (size_t)ROWS * 2 * DI;      // [ROWS, DI]
    float* xdbl = u    + (size_t)ROWS * DI;          // [ROWS, XD]
    float* dtf  = xdbl + (size_t)ROWS * XD;          // [ROWS, DI]
    float* ys   = dtf  + (size_t)ROWS * DI;          // [ROWS, DI]
    float* hbuf = ys   + (size_t)ROWS * DI;          // [ROWS, DM]
    float* x1   = hbuf + (size_t)ROWS * DM;          // [ROWS, DM]

    for (int l = 0; l < 2; ++l) {
        const float* xin  = (l == 0) ? x : x1;
        float*       xout = (l == 1) ? (float*)d_out : x1;

        // 1) in_proj: xz = xin @ in_w[l]^T   [2048 x 4096], K=1024
        gemm_rt_wmma<<<dim3(ROWS / TM, (2 * DI) / TN), 256, 0, stream>>>(
            xin, DM, in_w + (size_t)l * 2 * DI * DM, xz,
            ROWS, 2 * DI, DM, nullptr, 0);

        // 2) causal depthwise conv + silu -> u
        conv_silu_kernel<<<(ROWS * DI) / 256, 256, 0, stream>>>(
            xz, conv_w + (size_t)l * DI * DC, conv_b + (size_t)l * DI, u);

        // 3) x_proj: xdbl = u @ xp_w[l]^T    [2048 x 96], K=2048
        gemm_rt_wmma<<<dim3(ROWS / TM, (XD + TN - 1) / TN), 256, 0, stream>>>(
            u, DI, xp_w + (size_t)l * XD * DI, xdbl,
            ROWS, XD, DI, nullptr, 0);

        // 4) dt_proj + softplus: dtf = softplus(xdbl[:, :64] @ dt_w[l]^T + dt_b[l])
        gemm_rt_wmma<<<dim3(ROWS / TM, DI / TN), 256, 0, stream>>>(
            xdbl, XD, dt_w + (size_t)l * DI * DTR, dtf,
            ROWS, DI, DTR, dt_b + (size_t)l * DI, 1);

        // 5) selective scan (fused with +u*D and *silu(z))
        scan_kernel<<<(BB * DI * DS) / 256, 256, 0, stream>>>(
            dtf, xdbl, u, xz, A_log + (size_t)l * DI * DS,
            Dsk + (size_t)l * DI, ys);

        // 6) out_proj: hbuf = ys @ out_w[l]^T  [2048 x 1024], K=2048
        gemm_rt_wmma<<<dim3(ROWS / TM, DM / TN), 256, 0, stream>>>(
            ys, DI, out_w + (size_t)l * DM * DI, hbuf,
            ROWS, DM, DI, nullptr, 0);

        // 7) LayerNorm + residual
        ln_residual_kernel<<<ROWS, 256, 0, stream>>>(
            hbuf, xin, ln_w + (size_t)l * DM, ln_b + (size_t)l * DM, xout);
    }
}